// anticipation_mae_train_35605278884350
// MI455X (gfx1250) — compile-verified
//
#include <hip/hip_runtime.h>
#include <stdint.h>

// Flip to 0 if the gfx1250 assembler ever rejects the async-to-LDS mnemonics
// (round-1 compile confirmed it accepts them).
#ifndef USE_ASYNC_LDS
#define USE_ASYNC_LDS 1
#endif

typedef __attribute__((ext_vector_type(4))) float v4f;

#define HVAL   7500.0f
#define EVAL   750.0f          // 0.1 * H
#define N_DIM  64
#define T_DIM  65536
#define TPB    256             // 8 wave32 waves
#define SPLIT  16              // blocks per n  -> 1024 blocks total
#define CHUNK  (T_DIM / SPLIT) // 4096 timesteps per block
#define ITERS  (CHUNK / TPB)   // 16 pipeline stages
#define NBUF   3               // async LDS ring depth (2 stages in flight)
#define NSLOT  32              // accumulator slots per n (28 used, layout below)

// Per-channel slot layout (8 slots, c*8 + ...):
//  [0] sum |d| over ALL elements   [1] sum_o  [2] cnt_o
//  [3] sum_u  [4] cnt_u            [5] sum_e  [6] cnt_e   [7] pad
// in-bin derived at finalize: s_in = tot - s_o - s_u ; c_in = T - c_o - c_u.

__device__ __forceinline__ void accum_elem(v4f o, v4f t, float* acc) {
#pragma unroll
  for (int c = 0; c < 4; ++c) {
    float tv = t[c];
    tv = (tv > HVAL || tv < 0.0f) ? HVAL : tv;     // clamp OOR targets to h
    float d  = __builtin_fabsf(o[c] - tv);
    float mo = (tv == HVAL) ? 1.0f : 0.0f;
    float mu = (tv == 0.0f) ? 1.0f : 0.0f;
    float me = (tv > 0.0f && tv < EVAL) ? 1.0f : 0.0f;
    float* a = acc + c * 8;
    a[0] += d;
    a[1] = __builtin_fmaf(mo, d, a[1]);  a[2] += mo;
    a[3] = __builtin_fmaf(mu, d, a[3]);  a[4] += mu;
    a[5] = __builtin_fmaf(me, d, a[5]);  a[6] += me;
  }
}

#if USE_ASYNC_LDS
// Per-lane async global->LDS copy of 16 bytes (VDST = LDS byte address VGPR).
// Generic LDS pointers are {aperture_hi32, lds_offset32}, so the low 32 bits
// are exactly the LDS byte address the instruction wants.
__device__ __forceinline__ void async_copy_b128(const v4f* gsrc, const v4f* ldst) {
  unsigned           loff  = (unsigned)(uintptr_t)ldst;
  unsigned long long gaddr = (unsigned long long)(uintptr_t)gsrc;
  asm volatile("global_load_async_to_lds_b128 %0, %1, off"
               :: "v"(loff), "v"(gaddr) : "memory");
}
#endif

__global__ __launch_bounds__(TPB) void mae_init(float* __restrict__ ws) {
  int i = blockIdx.x * TPB + threadIdx.x;   // grid = 8 blocks -> 2048 slots
  ws[i] = 0.0f;
}

__global__ __launch_bounds__(TPB) void mae_partial(const float* __restrict__ out_raw,
                                                   const float* __restrict__ tgt_raw,
                                                   float* __restrict__ ws) {
  const v4f* outp = (const v4f*)out_raw;    // one float4 = all 4 channels of one t
  const v4f* tgt  = (const v4f*)tgt_raw;
  const int n    = blockIdx.x / SPLIT;
  const int ch   = blockIdx.x % SPLIT;
  const int tid  = threadIdx.x;
  const int base = n * T_DIM + ch * CHUNK;  // in float4 units

  float acc[NSLOT];
#pragma unroll
  for (int i = 0; i < NSLOT; ++i) acc[i] = 0.0f;

#if USE_ASYNC_LDS
  // 3-deep async ring: each lane streams only its own slot, so no barriers are
  // needed — ASYNCcnt (per wave, in-order completion) is the only sync. Two
  // stages (4 B128 loads, 2 KB/wave) stay in flight for latency tolerance.
  __shared__ v4f tile[NBUF][2][TPB];        // [buf][0=output,1=target][lane]
  async_copy_b128(outp + base + tid,       &tile[0][0][tid]);
  async_copy_b128(tgt  + base + tid,       &tile[0][1][tid]);
  async_copy_b128(outp + base + TPB + tid, &tile[1][0][tid]);
  async_copy_b128(tgt  + base + TPB + tid, &tile[1][1][tid]);
#pragma unroll
  for (int k = 0; k < ITERS; ++k) {
    if (k + 2 < ITERS) {
      const int nb = (k + 2) % NBUF;
      async_copy_b128(outp + base + (k + 2) * TPB + tid, &tile[nb][0][tid]);
      async_copy_b128(tgt  + base + (k + 2) * TPB + tid, &tile[nb][1][tid]);
    }
    // Loads complete in order: leave the newest stages pending, ensure stage k
    // has landed. rem = stages after k still outstanding.
    const int rem = ITERS - 1 - k;
    if (rem >= 2)      asm volatile("s_wait_asynccnt 4" ::: "memory");
    else if (rem == 1) asm volatile("s_wait_asynccnt 2" ::: "memory");
    else               asm volatile("s_wait_asynccnt 0" ::: "memory");
    v4f o = tile[k % NBUF][0][tid];
    v4f t = tile[k % NBUF][1][tid];
    accum_elem(o, t, acc);
  }
#else
#pragma unroll 4
  for (int k = 0; k < ITERS; ++k) {
    const int idx = base + k * TPB + tid;
    v4f o = __builtin_nontemporal_load(outp + idx);
    v4f t = __builtin_nontemporal_load(tgt + idx);
    accum_elem(o, t, acc);
  }
#endif

  // wave32 butterfly transpose-reduce: 16+8+4+2+1 = 31 shuffles/thread, after
  // which acc[0] holds the wave-wide total of accumulator index == lane.
  const int lane = tid & 31;
#pragma unroll
  for (int b = 16; b >= 1; b >>= 1) {
    const bool hi = (lane & b) != 0;
#pragma unroll
    for (int j = 0; j < 16; ++j) {
      if (j >= b) break;                    // live length shrinks with b
      float send = hi ? acc[j] : acc[j + b];
      float recv = __shfl_xor(send, b, 32);
      float keep = hi ? acc[j + b] : acc[j];
      acc[j] = keep + recv;
    }
  }
  // 32 lanes -> 32 distinct addresses: conflict-free f32 atomics into L2.
  atomicAdd(&ws[n * NSLOT + lane], acc[0]);
}

__global__ __launch_bounds__(TPB) void mae_final(const float* __restrict__ ws,
                                                 float* __restrict__ out) {
  const int tid = threadIdx.x;              // 256 threads == 64*4 (n,c) pairs
  const int n = tid >> 2, c = tid & 3;
  const float* p = ws + n * NSLOT + c * 8;
  const float tot = p[0], s_o = p[1], c_o = p[2], s_u = p[3], c_u = p[4],
              s_e = p[5], c_e = p[6];
  const float c_in = (float)T_DIM - c_o - c_u;
  const float s_in = tot - s_o - s_u;
  const float NANF = __builtin_nanf("");

  const float m_o  = (c_o  > 0.0f) ? s_o  / c_o  : NANF;
  const float m_in = (c_in > 0.0f) ? s_in / c_in : NANF;
  const float m_u  = (c_u  > 0.0f) ? s_u  / c_u  : NANF;
  const float m_e  = (c_e  > 0.0f) ? s_e  / c_e  : NANF;

  float s2 = 0.0f, n2 = 0.0f;
  if (m_u == m_u)   { s2 += m_u;  n2 += 1.0f; }
  if (m_in == m_in) { s2 += m_in; n2 += 1.0f; }
  const float in_comb = (n2 > 0.0f) ? s2 / n2 : NANF;

  float s3 = 0.0f, n3 = 0.0f;
  if (m_o == m_o)         { s3 += m_o;     n3 += 1.0f; }
  if (in_comb == in_comb) { s3 += in_comb; n3 += 1.0f; }
  const float w = (n3 > 0.0f) ? s3 / n3 : NANF;

  // NaN-counting mean over the 256 (n,c) pairs for each of the 4 outputs.
  float qv[4] = {w, in_comb, m_o, m_e};
  float sv[4], cv[4];
#pragma unroll
  for (int q = 0; q < 4; ++q) {
    const bool ok = (qv[q] == qv[q]);
    sv[q] = ok ? qv[q] : 0.0f;
    cv[q] = ok ? 1.0f : 0.0f;
  }
#pragma unroll
  for (int b = 16; b >= 1; b >>= 1) {
#pragma unroll
    for (int q = 0; q < 4; ++q) {
      sv[q] += __shfl_xor(sv[q], b, 32);
      cv[q] += __shfl_xor(cv[q], b, 32);
    }
  }
  __shared__ float red[8][8];
  const int lane = tid & 31, wave = tid >> 5;
  if (lane == 0) {
#pragma unroll
    for (int q = 0; q < 4; ++q) { red[wave][q] = sv[q]; red[wave][q + 4] = cv[q]; }
  }
  __syncthreads();
  if (tid == 0) {
#pragma unroll
    for (int q = 0; q < 4; ++q) {
      float S = 0.0f, C = 0.0f;
      for (int wv = 0; wv < 8; ++wv) { S += red[wv][q]; C += red[wv][q + 4]; }
      out[q] = (C > 0.0f) ? S / C : NANF;   // (wMAE, inMAE, oMAE, eMAE)
    }
  }
}

extern "C" void kernel_launch(void* const* d_in, const int* in_sizes, int n_in,
                              void* d_out, int out_size, void* d_ws, size_t ws_size,
                              hipStream_t stream) {
  (void)in_sizes; (void)n_in; (void)out_size; (void)ws_size;
  const float* outp = (const float*)d_in[0];   // output_rsd [64,65536,4] f32
  const float* tgt  = (const float*)d_in[1];   // target_rsd [64,65536,4] f32
  float* ws  = (float*)d_ws;                   // 64*32 f32 accumulators (8 KB)
  float* out = (float*)d_out;                  // 4 f32 scalars

  hipLaunchKernelGGL(mae_init,    dim3((N_DIM * NSLOT) / TPB), dim3(TPB), 0, stream, ws);
  hipLaunchKernelGGL(mae_partial, dim3(N_DIM * SPLIT),         dim3(TPB), 0, stream,
                     outp, tgt, ws);
  hipLaunchKernelGGL(mae_final,   dim3(1),                     dim3(TPB), 0, stream, ws, out);
}